// Item2VecModel_74509092651223
// MI455X (gfx1250) — compile-verified
//
#include <hip/hip_runtime.h>
#include <hip/hip_bf16.h>
#include <cstdint>
#include <cstddef>

typedef float v2f __attribute__((ext_vector_type(2)));
typedef float v8f __attribute__((ext_vector_type(8)));

#define DIM 128
#define TILE 16
#define WAVES_PER_BLOCK 8

// One wave per 16-sample tile. Scores computed as the diagonal of a 16x16
// fp32 WMMA product: A = 16 center rows (16x128), B = 16 gathered W_out rows
// transposed to 128x16, accumulated with 32 chained V_WMMA_F32_16X16X4_F32.
__global__ __launch_bounds__(256) void i2v_tile_kernel(
    const int* __restrict__ center, const int* __restrict__ pos,
    const int* __restrict__ neg, const float* __restrict__ W_in,
    const float* __restrict__ W_out, float* __restrict__ partial,
    int ntiles, int nneg)
{
  const int lane = threadIdx.x & 31;
  const int wave = threadIdx.x >> 5;
  const int tile = blockIdx.x * WAVES_PER_BLOCK + wave;
  if (tile >= ntiles) return;          // wave-uniform guard: EXEC stays all-1s

  const int t  = tile * TILE;
  const int n  = lane & 15;            // matrix column / batch-in-tile id
  const int hl = lane >> 4;            // half-wave: selects K pair
  const int b  = t + n;

  // ---- A fragments: center embeddings in WMMA 32-bit A layout ----
  // lane L, step s holds K = 4*s + 2*hl + {0,1} of row M = L%16
  const int rowA = center[b];
  const float* pa = W_in + (size_t)rowA * DIM;
  v2f a[DIM / 4];
  #pragma unroll
  for (int s = 0; s < DIM / 4; ++s)
    a[s] = *(const v2f*)(pa + 4 * s + 2 * hl);

  // diagonal ownership: lane L holds D[n][n] iff (n>>3)==hl, at acc index n&7
  const int  v    = n & 7;
  const bool mine = ((n >> 3) == hl);

  float lane_loss = 0.0f;

  for (int k = 0; k <= nneg; ++k) {     // k==0 -> positive, else negative k-1
    const int rowB = (k == 0) ? pos[b] : neg[(size_t)b * nneg + (k - 1)];
    const float* pb = W_out + (size_t)rowB * DIM;

    v2f bf[DIM / 4];
    #pragma unroll
    for (int s = 0; s < DIM / 4; ++s)
      bf[s] = *(const v2f*)(pb + 4 * s + 2 * hl);

    v8f acc = {0.f, 0.f, 0.f, 0.f, 0.f, 0.f, 0.f, 0.f};
    #pragma unroll
    for (int s = 0; s < DIM / 4; ++s)
      acc = __builtin_amdgcn_wmma_f32_16x16x4_f32(
          false, a[s], false, bf[s], (short)0, acc, false, false);

    // extract diagonal element with constant-index selects (no scratch)
    float score = acc[0];
    #pragma unroll
    for (int j = 1; j < 8; ++j) score = (v == j) ? acc[j] : score;

    const float x   = (k == 0) ? score : -score;
    const float sig = 1.0f / (1.0f + expf(-x));
    const float l   = -logf(sig + 1e-10f);
    lane_loss += mine ? l : 0.0f;       // select, not branch: EXEC uniform
  }

  // fixed-order wave reduction -> deterministic across replays
  #pragma unroll
  for (int off = 16; off > 0; off >>= 1)
    lane_loss += __shfl_down(lane_loss, off, 32);
  if (lane == 0) partial[tile] = lane_loss;
}

// Deterministic single-block tree reduction; overwrites d_out (poison-safe).
__global__ __launch_bounds__(256) void i2v_reduce_kernel(
    const float* __restrict__ partial, float* __restrict__ out,
    int ntiles, float scale)
{
  __shared__ float sm[256];
  float s = 0.0f;
  for (int i = threadIdx.x; i < ntiles; i += 256) s += partial[i];
  sm[threadIdx.x] = s;
  __syncthreads();
  #pragma unroll
  for (int off = 128; off > 0; off >>= 1) {
    if ((int)threadIdx.x < off) sm[threadIdx.x] += sm[threadIdx.x + off];
    __syncthreads();
  }
  if (threadIdx.x == 0) out[0] = sm[0] * scale;
}

extern "C" void kernel_launch(void* const* d_in, const int* in_sizes, int n_in,
                              void* d_out, int out_size, void* d_ws, size_t ws_size,
                              hipStream_t stream) {
  const int*   center = (const int*)d_in[0];
  const int*   pos    = (const int*)d_in[1];
  const int*   neg    = (const int*)d_in[2];
  const float* W_in   = (const float*)d_in[3];
  const float* W_out  = (const float*)d_in[4];
  float* out = (float*)d_out;
  float* partial = (float*)d_ws;

  const int nb     = in_sizes[0];           // B = 16384
  const int nneg   = in_sizes[2] / nb;      // 20
  const int ntiles = nb / TILE;             // 1024

  const int blocks = (ntiles + WAVES_PER_BLOCK - 1) / WAVES_PER_BLOCK;
  i2v_tile_kernel<<<blocks, 256, 0, stream>>>(center, pos, neg, W_in, W_out,
                                              partial, ntiles, nneg);
  i2v_reduce_kernel<<<1, 256, 0, stream>>>(partial, out, ntiles,
                                           1.0f / (float)nb);
}